// SACSMA_472446402987
// MI455X (gfx1250) — compile-verified
//
#include <hip/hip_runtime.h>
#include <hip/hip_bf16.h>

// SAC-SMA / HBV scan for MI455X (gfx1250, wave32).
// One thread per (grid, mu); serial 365-step scan in registers.
// Dynamic params (idx 0..4) 4-deep double-buffered through LDS via
// global_load_async_to_lds_b32 (ASYNCcnt path, NT temporal hint);
// static params (idx 5..10) read once from step NSTEP-1.
// mu-mean via wave32 shfl_xor over the 8-lane mu group.
// v_rcp_f32 fast reciprocals replace IEEE divide expansions (latency-bound
// loop: only 1000 wave32s exist for the whole chip).

#define NSTEP 365
#define NGRID 4000
#define MU    8
#define GPB   8               // grids per block
#define TPB   (GPB * MU)      // 64 threads = 2 wave32s
#define PF    4               // prefetch pipeline depth (LDS buffers)
#define SLOT  8               // floats per thread slot (5 used, padded to 32B)
#define SMIN   1.0e-4f
#define CAPMIN 1.25e-3f       // 0.005/4

__device__ __forceinline__ float rcpf(float x) { return __builtin_amdgcn_rcpf(x); }

// Issue 5 async B32 copies, i=0..4: the float at byte offset goff + 32*i in gbase goes to LDS byte address ldsoff + 4*i
__device__ __forceinline__ void async_copy5(const float* __restrict__ gbase,
                                            unsigned goff, unsigned ldsoff)
{
#pragma unroll
    for (int i = 0; i < 5; ++i) {
        asm volatile("global_load_async_to_lds_b32 %0, %1, %2 th:TH_LOAD_NT"
                     :
                     : "v"(ldsoff + 4u * (unsigned)i),
                       "v"(goff + (unsigned)(4 * MU) * (unsigned)i),
                       "s"(gbase)
                     : "memory");
    }
}

__device__ __forceinline__ unsigned dyn_goff(int t, int g, int m)
{
    // byte offset of parameters[t, g, 0, m]; array is ~514MB < 2^31 so a
    // 32-bit VGPR offset against the SGPR base (GVS mode) is safe.
    return (((unsigned)t * (unsigned)NGRID + (unsigned)g) * (11u * MU)
            + (unsigned)m) * 4u;
}

__global__ __launch_bounds__(TPB)
void sacsma_scan_kernel(const float* __restrict__ x,      // (365,4000,3)
                        const float* __restrict__ par,    // (365,4000,11,8)
                        float* __restrict__ out)          // (365,4000,2)
{
    __shared__ float sdyn[PF][TPB * SLOT];

    const int tid = threadIdx.x;
    const int m   = tid & (MU - 1);
    const int g   = blockIdx.x * GPB + (tid >> 3);

    // LDS byte offset of this thread's slot in buffer 0; buffers are
    // TPB*SLOT*4 bytes apart. (the LDS address is the low 32 bits of the
    // generic pointer to a __shared__ object)
    const unsigned ldsT   = (unsigned)(size_t)(&sdyn[0][tid * SLOT]);
    const unsigned ldsStr = (unsigned)(TPB * SLOT * 4);

    // ---- static parameters, taken from t = NSTEP-1 (STAIND) ----
    const size_t sb = (((size_t)(NSTEP - 1) * NGRID + g) * 11) * MU + m;
    const float rexp  = 7.0f * par[sb + 5 * MU];
    const float f3    = 0.005f + 0.99f * par[sb + 6 * MU];
    const float f4    = 0.005f + 0.99f * par[sb + 7 * MU];
    const float pfree = par[sb + 8 * MU];
    const float klzp  = par[sb + 9 * MU];
    const float klzs  = par[sb + 10 * MU];
    const float erex  = 1.0f + rexp;

    float S1 = 0.001f, S2 = 0.001f, S3 = 0.001f, S4 = 0.001f, S5 = 0.001f;

    // Prologue: prefetch steps 0..PF-1 (PF*5 = 20 outstanding async ops).
#pragma unroll
    for (int t0 = 0; t0 < PF; ++t0)
        async_copy5(par, dyn_goff(t0, g, m), ldsT + (unsigned)t0 * ldsStr);

#pragma unroll 1
    for (int t = 0; t < NSTEP; ++t) {
        const int b = t & (PF - 1);

        // Async loads complete in issue order: <= (PF-1)*5 outstanding means
        // the oldest (current) buffer's 5 copies have landed in LDS.
        asm volatile("s_wait_asynccnt 0xf" ::: "memory");

        const float* s = &sdyn[b][tid * SLOT];
        const float pctim = s[0];
        const float smax  = 1.0f + 1999.0f * s[1];
        const float f1    = 0.005f + 0.99f * s[2];
        const float f2    = 0.005f + 0.99f * s[3];
        const float kuz   = s[4];

        // LDS reads must retire before the async engine may overwrite this
        // buffer (WAR on LDS), then prefetch step t+PF into it.
        asm volatile("s_wait_dscnt 0x0" ::: "memory");
        int tn = t + PF; if (tn > NSTEP - 1) tn = NSTEP - 1;
        async_copy5(par, dyn_goff(tn, g, m), ldsT + (unsigned)b * ldsStr);

        const size_t xo = ((size_t)t * NGRID + g) * 3;
        const float P  = x[xo + 0];
        const float Ep = x[xo + 2];

        // ---- derived capacities ----
        const float uztwm  = f1 * smax;
        const float rest1  = smax - uztwm;
        const float uzfwm  = fmaxf(CAPMIN, f2 * rest1);
        const float rest2  = rest1 - uzfwm;
        const float lztwm  = fmaxf(CAPMIN, f3 * rest2);
        const float rest3  = rest2 - lztwm;
        const float lzfwpm = fmaxf(CAPMIN, f4 * rest3);
        const float lzfwsm = fmaxf(CAPMIN, (1.0f - f4) * rest3);
        const float pbase  = lzfwpm * klzp + lzfwsm * klzs;
        const float rden   = rcpf(pbase);
        const float zperc  = fminf(100000.0f,
            (lztwm + lzfwsm * (1.0f - klzs)) * rden + lzfwpm * (1.0f - klzp) * rden);

        // ---- fluxes ----
        const float qdir = pctim * P;
        const float peff = (1.0f - pctim) * P;

        const float r_uztwm = rcpf(uztwm);
        const float r_uzfwm = rcpf(uzfwm);
        const float ru = (S1 * r_uztwm < S2 * r_uzfwm)
                           ? (S2 * uztwm - S1 * uzfwm) * rcpf(uztwm + uzfwm)
                           : 0.0f;
        const float euztw = fminf(S1 * r_uztwm * Ep, S1);
        const float twexu = (S1 >= uztwm) ? peff  : 0.0f;
        const float qsur  = (S2 >= uzfwm) ? twexu : 0.0f;
        const float qint  = kuz * S2;
        const float euzfw = fminf(fmaxf(0.0f, Ep - euztw), S2);

        const float deficit = fmaxf(1e-8f, lztwm - S3)
                            + fmaxf(1e-8f, lzfwpm - S4)
                            + fmaxf(1e-8f, lzfwsm - S5);
        const float defmax  = fmaxf(1e-8f, lztwm + lzfwpm + lzfwsm);
        float pc = pbase * (1.0f + zperc * __powf(deficit * rcpf(defmax), erex))
                   * S2 * r_uzfwm;
        pc = fmaxf(0.0f, fminf(pc, S2));

        const float pctw    = (1.0f - pfree) * pc;
        const float r_lztwm = rcpf(lztwm);
        const float elztw   = fminf(S3 * r_lztwm * fmaxf(0.0f, Ep - euztw - euzfw), S3);
        const float twexl   = (S3 >= lztwm) ? pctw : 0.0f;

        const float d1  = fmaxf(0.0f, 1.0f - S4 * rcpf(lzfwpm));
        const float d2  = fmaxf(0.0f, 1.0f - S5 * rcpf(lzfwsm));
        const float tot = d1 + d2;
        const float rps = rcpf(lzfwpm + lzfwsm);
        // fp + fs == 1 in both branches of deficit_dist
        const float fp  = (tot > 0.0f) ? d1 * rcpf(tot) : lzfwpm * rps;
        const float fs  = 1.0f - fp;

        const float twexlp = fp * twexl;
        const float twexls = fs * twexl;
        const float pcfwp  = pfree * fp * pc;
        const float pcfws  = pfree * fs * pc;

        const float rsum  = rcpf(lztwm + lzfwpm + lzfwsm);
        const bool  maskL = (S3 * r_lztwm) < ((S4 + S5) * rps);
        const float rlp   = maskL ? (S4 * lztwm - S3 * lzfwpm) * rsum : 0.0f;
        const float rls   = maskL ? (S5 * lztwm - S3 * lzfwsm) * rsum : 0.0f;

        const float qbfp = klzp * S4;
        const float qbfs = klzs * S5;

        // ---- state update ----
        S1 = fmaxf(S1 + peff + ru - euztw - twexu, SMIN);
        S2 = fmaxf(S2 + twexu - euzfw - qsur - qint - ru - pc, SMIN);
        S3 = fmaxf(S3 + pctw + rlp + rls - elztw - twexl, SMIN);
        S4 = fmaxf(S4 + twexlp + pcfwp - rlp - qbfp, SMIN);
        S5 = fmaxf(S5 + twexls + pcfws - rls - qbfs, SMIN);

        float qsim = qdir + qsur + qint + qbfp + qbfs;
        float et   = euztw + euzfw + elztw;

        // ---- mean over MU=8 lanes (wave32 xor-shuffle tree) ----
        qsim += __shfl_xor(qsim, 1, 32);
        qsim += __shfl_xor(qsim, 2, 32);
        qsim += __shfl_xor(qsim, 4, 32);
        et   += __shfl_xor(et,   1, 32);
        et   += __shfl_xor(et,   2, 32);
        et   += __shfl_xor(et,   4, 32);

        if (m == 0) {
            const size_t oo = ((size_t)t * NGRID + g) * 2;
            __builtin_nontemporal_store(qsim * 0.125f, &out[oo + 0]);
            __builtin_nontemporal_store(et   * 0.125f, &out[oo + 1]);
        }
    }
}

extern "C" void kernel_launch(void* const* d_in, const int* in_sizes, int n_in,
                              void* d_out, int out_size, void* d_ws, size_t ws_size,
                              hipStream_t stream)
{
    const float* x   = (const float*)d_in[0];   // (365,4000,3)  f32
    const float* par = (const float*)d_in[1];   // (365,4000,11,8) f32
    float* out = (float*)d_out;                 // (365,4000,2)  f32

    dim3 grid(NGRID / GPB);                     // 500 blocks
    dim3 block(TPB);                            // 64 threads (2 wave32s)
    sacsma_scan_kernel<<<grid, block, 0, stream>>>(x, par, out);
}